// ActivationGATLayer_83476984365545
// MI455X (gfx1250) — compile-verified
//
#include <hip/hip_runtime.h>

typedef __attribute__((ext_vector_type(2))) float v2f;
typedef __attribute__((ext_vector_type(8))) float v8f;

#define DIM 64
#define NHEADS 4

// ---------------- zero scratch ----------------
__global__ __launch_bounds__(256) void k_zero(float* __restrict__ p, long long n) {
  long long i = (long long)blockIdx.x * blockDim.x + threadIdx.x;
  if (i < n) p[i] = 0.0f;
}

// ---------------- z = h * norm ----------------
__global__ __launch_bounds__(256) void k_compute_z(const float* __restrict__ h,
                                                   const float* __restrict__ norm,
                                                   float* __restrict__ z, long long total) {
  long long i = (long long)blockIdx.x * blockDim.x + threadIdx.x;
  if (i < total) z[i] = h[i] * norm[i >> 6];  // norm is [N,1]
}

// ---------------- edge scores via V_WMMA_F32_16X16X4_F32 ----------------
// One wave computes 16 edge scores: C(16x16) = Zsrc(16x64) * Zdst(16x64)^T
// accumulated over 16 WMMA steps of K=4; diagonal C[i][i] = dot(z_src_i, z_dst_i).
// A 16x4 f32 layout: lane l (m=l&15, khalf=l>>4): v[0]=A[m][2*khalf], v[1]=A[m][2*khalf+1].
// B 4x16 f32 layout mirrors it with rows<->K: lane l: v[0]=B[2*khalf][m], v[1]=B[2*khalf+1][m].
// Both reduce to the same per-lane float2 gather from row `edge_m`, cols {4k+2*khalf, +1}.
__global__ __launch_bounds__(256) void k_edge_scores(
    const float* __restrict__ z, const int* __restrict__ src, const int* __restrict__ dst,
    float* __restrict__ scores, unsigned int* __restrict__ smax, int nEdges) {
  const int lane = threadIdx.x & 31;
  const int wave = (int)(((unsigned)blockIdx.x * blockDim.x + threadIdx.x) >> 5);
  const int base = wave * 16;
  if (base >= nEdges) return;  // wave-uniform

  const int m = lane & 15;      // which of the 16 edges this lane serves
  const int khalf = lane >> 4;  // column-pair select (K offset 0 or 2)
  const int eReal = base + m;
  const int e = (eReal < nEdges) ? eReal : (nEdges - 1);  // branchless clamp: keep EXEC full
  const int di = dst[e];
  const long long rs = (long long)src[e] * DIM;
  const long long rd = (long long)di * DIM;

  v8f c = {};
#pragma unroll
  for (int kc = 0; kc < 16; ++kc) {
    const int col = kc * 4 + khalf * 2;
    v2f a = *(const v2f*)(z + rs + col);
    v2f b = *(const v2f*)(z + rd + col);
    c = __builtin_amdgcn_wmma_f32_16x16x4_f32(false, a, false, b, (short)0, c, false, false);
  }

  // Diagonal extraction from C/D layout:
  //   edge i<8  lives at (VGPR i,   lane i)
  //   edge i>=8 lives at (VGPR i-8, lane 16+i)  -> lanes 24..31 hold edges 8..15
  int idx = -1;
  if (lane < 8) idx = lane;            // edge = lane       (== m)
  else if (lane >= 24) idx = lane - 24;  // edge = lane - 16 (== m)
  if (idx >= 0 && eReal < nEdges) {
    float s = c[0];
#pragma unroll
    for (int r = 1; r < 8; ++r)
      if (idx == r) s = c[r];
    s = fmaxf(s, 0.0f);  // relu => s >= 0
    scores[eReal] = s;
    // non-negative floats: IEEE order == unsigned bit-pattern order; init 0 == empty-seg fixup
    atomicMax(&smax[di], __float_as_uint(s));
  }
}

// ---------------- ex = exp(s - max); denom += ex ----------------
__global__ __launch_bounds__(256) void k_edge_exp(float* __restrict__ scores,
                                                  const int* __restrict__ dst,
                                                  const unsigned int* __restrict__ smax,
                                                  float* __restrict__ ssum, int nEdges) {
  int e = blockIdx.x * blockDim.x + threadIdx.x;
  if (e >= nEdges) return;
  int d = dst[e];
  float mx = __uint_as_float(smax[d]);
  float ex = expf(scores[e] - mx);
  scores[e] = ex;  // overwrite in place
  atomicAdd(&ssum[d], ex);
}

// ---------------- agg[dst] += (ex/denom[dst]) * z[src] ----------------
// One wave per edge; lane handles 2 columns (float2 load, 2 f32 atomics).
__global__ __launch_bounds__(256) void k_aggregate(const float* __restrict__ z,
                                                   const int* __restrict__ src,
                                                   const int* __restrict__ dst,
                                                   const float* __restrict__ ex,
                                                   const float* __restrict__ ssum,
                                                   float* __restrict__ agg, int nEdges) {
  const int lane = threadIdx.x & 31;
  const int e = (int)(((unsigned)blockIdx.x * blockDim.x + threadIdx.x) >> 5);
  if (e >= nEdges) return;
  const int s = src[e], d = dst[e];
  const float alpha = ex[e] / ssum[d];
  const v2f zv = *(const v2f*)(z + (long long)s * DIM + lane * 2);
  float* ar = agg + (long long)d * DIM + lane * 2;
  atomicAdd(&ar[0], alpha * zv.x);
  atomicAdd(&ar[1], alpha * zv.y);
}

// ---------------- BN column stats: colsum, colsumsq ----------------
__global__ __launch_bounds__(256) void k_bn_stats(const float* __restrict__ agg,
                                                  float* __restrict__ colsum,
                                                  float* __restrict__ colsumsq, int nNodes) {
  __shared__ float sh_s[256];
  __shared__ float sh_q[256];
  const int d = threadIdx.x & 63;
  const int g = threadIdx.x >> 6;
  float s = 0.f, q = 0.f;
#pragma unroll 4
  for (int k = 0; k < 32; ++k) {
    int row = blockIdx.x * 128 + k * 4 + g;
    if (row < nNodes) {
      float v = agg[(long long)row * DIM + d];
      s += v;
      q += v * v;
    }
  }
  sh_s[threadIdx.x] = s;
  sh_q[threadIdx.x] = q;
  __syncthreads();
  if (g == 0) {
    s = sh_s[d] + sh_s[64 + d] + sh_s[128 + d] + sh_s[192 + d];
    q = sh_q[d] + sh_q[64 + d] + sh_q[128 + d] + sh_q[192 + d];
    atomicAdd(&colsum[d], s);
    atomicAdd(&colsumsq[d], q);
  }
}

// ---------------- BN normalize + per-head affine + relu + *norm ----------------
__global__ __launch_bounds__(256) void k_finalize(const float* __restrict__ agg,
                                                  const float* __restrict__ colsum,
                                                  const float* __restrict__ colsumsq,
                                                  const float* __restrict__ gamma,
                                                  const float* __restrict__ beta,
                                                  const float* __restrict__ norm,
                                                  float* __restrict__ out, int nNodes) {
  long long gid = (long long)blockIdx.x * blockDim.x + threadIdx.x;
  long long total = (long long)nNodes * DIM;
  if (gid >= total) return;
  int d = (int)(gid & 63);
  long long n = gid >> 6;
  float invN = 1.0f / (float)nNodes;
  float mean = colsum[d] * invN;
  float var = colsumsq[d] * invN - mean * mean;  // biased, matches jnp.var
  float inv = rsqrtf(var + 1e-5f);
  float xh = (agg[gid] - mean) * inv;
  float nr = norm[n];
#pragma unroll
  for (int hd = 0; hd < NHEADS; ++hd) {
    float v = fmaxf(gamma[hd * DIM + d] * xh + beta[hd * DIM + d], 0.0f) * nr;
    out[n * (NHEADS * DIM) + hd * DIM + d] = v;
  }
}

extern "C" void kernel_launch(void* const* d_in, const int* in_sizes, int n_in,
                              void* d_out, int out_size, void* d_ws, size_t ws_size,
                              hipStream_t stream) {
  const float* h = (const float*)d_in[0];
  const float* e = (const float*)d_in[1];
  const float* norm = (const float*)d_in[2];
  const float* gamma = (const float*)d_in[3];
  const float* beta = (const float*)d_in[4];
  const int* src = (const int*)d_in[5];
  const int* dst = (const int*)d_in[6];
  const int N = in_sizes[0] / DIM;
  const int E = in_sizes[5];

  // workspace layout (floats)
  float* wsf = (float*)d_ws;
  size_t zOff = 0;
  size_t sOff = zOff + (size_t)N * DIM;   // scores / ex (E)
  size_t maxOff = sOff + (size_t)E;       // smax (N)
  size_t sumOff = maxOff + (size_t)N;     // ssum (N)
  size_t aggOff = sumOff + (size_t)N;     // agg (N*D)
  size_t csOff = aggOff + (size_t)N * DIM;
  size_t csqOff = csOff + DIM;
  size_t endOff = csqOff + DIM;

  float* z = wsf + zOff;
  float* scores = wsf + sOff;
  unsigned int* smax = (unsigned int*)(wsf + maxOff);
  float* ssum = wsf + sumOff;
  float* agg = wsf + aggOff;
  float* colsum = wsf + csOff;
  float* colsumsq = wsf + csqOff;

  // 1) zero [smax .. colsumsq]
  long long zeroLen = (long long)(endOff - maxOff);
  k_zero<<<(int)((zeroLen + 255) / 256), 256, 0, stream>>>(wsf + maxOff, zeroLen);

  // 2) z = h * norm
  long long zTotal = (long long)N * DIM;
  k_compute_z<<<(int)((zTotal + 255) / 256), 256, 0, stream>>>(h, norm, z, zTotal);

  // 3) edge scores (WMMA) + segment max
  int waves = (E + 15) / 16;
  long long wthreads = (long long)waves * 32;
  k_edge_scores<<<(int)((wthreads + 255) / 256), 256, 0, stream>>>(z, src, dst, scores, smax, E);

  // 4) exp + segment sum
  k_edge_exp<<<(E + 255) / 256, 256, 0, stream>>>(scores, dst, smax, ssum, E);

  // 5) weighted scatter-sum aggregation
  long long athreads = (long long)E * 32;
  k_aggregate<<<(int)((athreads + 255) / 256), 256, 0, stream>>>(z, src, dst, scores, ssum, agg, E);

  // 6) BN column stats
  k_bn_stats<<<(N + 127) / 128, 256, 0, stream>>>(agg, colsum, colsumsq, N);

  // 7) BN + per-head affine + relu + norm -> h_out [N, H*D]
  k_finalize<<<(int)((zTotal + 255) / 256), 256, 0, stream>>>(agg, colsum, colsumsq, gamma, beta,
                                                              norm, (float*)d_out, N);

  // 8) e passthrough
  hipMemcpyAsync((float*)d_out + (size_t)N * DIM * NHEADS, e, (size_t)E * DIM * sizeof(float),
                 hipMemcpyDeviceToDevice, stream);
}